// SageLayer_27831388078277
// MI455X (gfx1250) — compile-verified
//
#include <hip/hip_runtime.h>

// GraphSAGE layer for MI455X (gfx1250, wave32).
// out = h@W_self + b_self + agg@W_neigh + b_neigh,
// agg = (scatter_sum(h[src]->dst) + h) / (deg + 1)   [self-loop folded in]
//
// Bottleneck: edge scatter (~0.3 GB of f32 atomics into L2). GEMM is only
// 3.3 GFLOP -> use the full-precision CDNA5 f32 tensor path
// (v_wmma_f32_16x16x4_f32); no reason to downcast.
//
// LDS layout for weights is k-pair interleaved (sW[k/2][n][k&1], padded row
// stride) so each WMMA B-fragment is a single aligned ds_load_b64 into an
// even VGPR pair -> no register repacking in front of the WMMAs, and the two
// half-wave pair-rows map to disjoint LDS bank halves (stride 288 % 64 == 32).

#define N_NODES 50000
#define DIM     128
#define N_EDGES 600000
#define M_TILES ((N_NODES + 15) / 16)   // 3125 (exact: 3125*16 = 50000)

#define PR_STRIDE 288                    // dwords per pair-row (128*2 + 32 pad)

typedef __attribute__((ext_vector_type(2))) float v2f;
typedef __attribute__((ext_vector_type(8))) float v8f;

// ---------------------------------------------------------------------------
// Kernel 1: edge scatter. One wave per edge: gather 128 f32 of h[src]
// (16B per lane), atomic-add into msg_sum[dst]; lane 0 bumps degree.
// ---------------------------------------------------------------------------
__global__ __launch_bounds__(256)
void sage_scatter(const float* __restrict__ h,
                  const int* __restrict__ edges,
                  float* __restrict__ msg,
                  float* __restrict__ deg) {
    const int wave = (int)((blockIdx.x * blockDim.x + threadIdx.x) >> 5); // edge id
    const int lane = threadIdx.x & 31;
    if (wave >= N_EDGES) return;                 // wave-uniform exit

    const int src = edges[wave];                 // edges[0][e]
    const int dst = edges[N_EDGES + wave];       // edges[1][e]

    const float4 v = ((const float4*)(h + (size_t)src * DIM))[lane];
    float* m = msg + (size_t)dst * DIM + lane * 4;
    atomicAdd(m + 0, v.x);
    atomicAdd(m + 1, v.y);
    atomicAdd(m + 2, v.z);
    atomicAdd(m + 3, v.w);
    if (lane == 0) atomicAdd(deg + dst, 1.0f);
}

// ---------------------------------------------------------------------------
// Kernel 2: fold self loop + mean-normalize, in place:
//   msg[n][k] = (msg[n][k] + h[n][k]) / (deg[n] + 1)
// ---------------------------------------------------------------------------
__global__ __launch_bounds__(256)
void sage_norm(const float* __restrict__ h,
               const float* __restrict__ deg,
               float* __restrict__ msg) {
    const size_t idx = (size_t)blockIdx.x * blockDim.x + threadIdx.x;
    if (idx >= (size_t)N_NODES * DIM) return;
    const int n = (int)(idx >> 7);
    msg[idx] = (msg[idx] + h[idx]) / (deg[n] + 1.0f);
}

// ---------------------------------------------------------------------------
// Cooperative stage of a 128x128 row-major weight matrix into the pair-
// interleaved LDS layout: sW[k>>1][n][k&1] with padded pair-row stride.
// ---------------------------------------------------------------------------
__device__ __forceinline__
void stage_weights(float* __restrict__ sW, const float* __restrict__ W, int tid) {
    for (int i = tid; i < DIM * DIM / 4; i += 256) {
        const int k = i >> 5;                    // source row
        const int c = (i & 31) * 4;              // source col (float4)
        const float4 v = ((const float4*)W)[i];
        float* d = sW + (k >> 1) * PR_STRIDE + (k & 1);
        d[(c + 0) * 2] = v.x;
        d[(c + 1) * 2] = v.y;
        d[(c + 2) * 2] = v.z;
        d[(c + 3) * 2] = v.w;
    }
}

// ---------------------------------------------------------------------------
// Kernel 3: fused dual GEMM with v_wmma_f32_16x16x4_f32.
// Block = 256 threads = 8 waves; each wave owns one 16-node M-tile and
// produces a full 16x128 output stripe (8 accumulator tiles, 64 VGPRs).
// Phase 1: W_self staged in LDS, accumulate h @ W_self.
// Phase 2: W_neigh staged in the same LDS, accumulate agg @ W_neigh.
// Bias added at store.
// ---------------------------------------------------------------------------
__global__ __launch_bounds__(256)
void sage_gemm(const float* __restrict__ h,
               const float* __restrict__ agg,
               const float* __restrict__ Wself,
               const float* __restrict__ bself,
               const float* __restrict__ Wneigh,
               const float* __restrict__ bneigh,
               float* __restrict__ out) {
    __shared__ float sW[64 * PR_STRIDE];   // 72 KB (of 320 KB/WGP)

    const int tid  = threadIdx.x;
    const int lane = tid & 31;
    const int wave = tid >> 5;

    stage_weights(sW, Wself, tid);
    __syncthreads();

    const int mtile   = blockIdx.x * 8 + wave;
    const bool active = (mtile < M_TILES);       // wave-uniform -> EXEC all-1s
    const int m0   = mtile * 16;
    const int col  = lane & 15;                  // N column / A row within tile
    const int half = lane >> 4;                  // 0: K 0,1   1: K 2,3
    const int koff = half * 2;

    // Per-lane base into pair-interleaved LDS: pair-row 'half', column 'col'.
    const float* bbase = sW + half * PR_STRIDE + col * 2;

    v8f acc[8] = {};                             // 8 N-tiles x 8 VGPRs

    if (active) {
        const float* aptr = h + (size_t)(m0 + col) * DIM + koff;
        for (int kt = 0; kt < 32; ++kt) {
            const v2f a = *(const v2f*)(aptr + kt * 4);      // A frag (b64)
            const float* bk = bbase + kt * (2 * PR_STRIDE);
            #pragma unroll
            for (int nt = 0; nt < 8; ++nt) {
                const v2f b = *(const v2f*)(bk + nt * 32);   // B frag (ds b64)
                acc[nt] = __builtin_amdgcn_wmma_f32_16x16x4_f32(
                    false, a, false, b, (short)0, acc[nt], false, false);
            }
        }
    }
    __syncthreads();

    stage_weights(sW, Wneigh, tid);
    __syncthreads();

    if (active) {
        const float* aptr = agg + (size_t)(m0 + col) * DIM + koff;
        for (int kt = 0; kt < 32; ++kt) {
            const v2f a = *(const v2f*)(aptr + kt * 4);
            const float* bk = bbase + kt * (2 * PR_STRIDE);
            #pragma unroll
            for (int nt = 0; nt < 8; ++nt) {
                const v2f b = *(const v2f*)(bk + nt * 32);
                acc[nt] = __builtin_amdgcn_wmma_f32_16x16x4_f32(
                    false, a, false, b, (short)0, acc[nt], false, false);
            }
        }

        // Store: C/D layout -> VGPR r holds rows m0+r (lanes 0-15) and
        // m0+8+r (lanes 16-31), column n0+col.
        const int rbase = m0 + half * 8;
        #pragma unroll
        for (int nt = 0; nt < 8; ++nt) {
            const int n0 = nt * 16;
            const float bias = bself[n0 + col] + bneigh[n0 + col];
            #pragma unroll
            for (int r = 0; r < 8; ++r) {
                out[(size_t)(rbase + r) * DIM + n0 + col] = acc[nt][r] + bias;
            }
        }
    }
}

// ---------------------------------------------------------------------------
extern "C" void kernel_launch(void* const* d_in, const int* in_sizes, int n_in,
                              void* d_out, int out_size, void* d_ws, size_t ws_size,
                              hipStream_t stream) {
    const float* h      = (const float*)d_in[0];
    const int*   edges  = (const int*)  d_in[1];
    const float* Wself  = (const float*)d_in[2];
    const float* bself  = (const float*)d_in[3];
    const float* Wneigh = (const float*)d_in[4];
    const float* bneigh = (const float*)d_in[5];
    float* out = (float*)d_out;

    float* msg = (float*)d_ws;                        // [N_NODES * DIM]
    float* deg = msg + (size_t)N_NODES * DIM;         // [N_NODES]

    // Zero scratch every call (graph-capture legal, deterministic).
    hipMemsetAsync(d_ws, 0,
                   ((size_t)N_NODES * DIM + N_NODES) * sizeof(float), stream);

    // 1 wave per edge -> 8 edges per 256-thread block.
    sage_scatter<<<(N_EDGES + 7) / 8, 256, 0, stream>>>(h, edges, msg, deg);

    sage_norm<<<((size_t)N_NODES * DIM + 255) / 256, 256, 0, stream>>>(h, deg, msg);

    // 3125 M-tiles, 8 waves (tiles) per block.
    sage_gemm<<<(M_TILES + 7) / 8, 256, 0, stream>>>(h, msg, Wself, bself,
                                                     Wneigh, bneigh, out);
}